// TopKActivationSparsifier_8392366096616
// MI455X (gfx1250) — compile-verified
//
#include <hip/hip_runtime.h>
#include <stdint.h>

#define DDIM 2048
#define TPB  256
#define EPT  (DDIM / TPB)  // 8 elements per thread

// Order-preserving map fp32 -> u32 (larger key <=> larger float)
__device__ __forceinline__ uint32_t f2key(float f) {
    uint32_t u = __float_as_uint(f);
    return (u & 0x80000000u) ? ~u : (u | 0x80000000u);
}

__global__ __launch_bounds__(TPB)
void topk_sparsify_kernel(const float* __restrict__ x,
                          const int* __restrict__ kp,
                          float* __restrict__ out) {
    __shared__ float    sx[DDIM];
    __shared__ uint32_t hist[TPB];   // 256 radix bins / per-thread tie counts
    __shared__ uint32_t pfx[TPB];    // per-thread exclusive tie prefix
    __shared__ uint32_t s_digit;
    __shared__ uint32_t s_rem;

    const int t   = threadIdx.x;
    const int row = blockIdx.x;
    const float* rowp = x   + (size_t)row * DDIM;
    float*       orow = out + (size_t)row * DDIM;

    // ---- Stage row into LDS via gfx1250 async global->LDS (ASYNCcnt) ----
    {
        uint32_t lds0 = (uint32_t)(uintptr_t)(&sx[0]) + (uint32_t)(t * 16);
        uint64_t g0   = (uint64_t)(uintptr_t)rowp + (uint64_t)(t * 16);
        asm volatile("global_load_async_to_lds_b128 %0, %1, off"
                     :: "v"(lds0), "v"(g0) : "memory");
        asm volatile("global_load_async_to_lds_b128 %0, %1, off offset:4096"
                     :: "v"(lds0), "v"(g0) : "memory");
        asm volatile("s_wait_asynccnt 0" ::: "memory");
    }
    __syncthreads();

    // ---- Pull my contiguous 8-element chunk into registers (2x ds_load_b128) ----
    const int base = t * EPT;
    float4 fa = *(const float4*)&sx[base];
    float4 fb = *(const float4*)&sx[base + 4];
    float    f[EPT]   = {fa.x, fa.y, fa.z, fa.w, fb.x, fb.y, fb.z, fb.w};
    uint32_t key[EPT];
    #pragma unroll
    for (int j = 0; j < EPT; ++j) key[j] = f2key(f[j]);

    const uint32_t k = (uint32_t)kp[0];
    uint32_t rem     = k;   // still to place at current radix level
    uint32_t kprefix = 0;   // decided high bits of the k-th largest key
    int      kbits   = 0;

    // ---- 4-pass radix select (8 bits/pass, MSB first), keys in registers ----
    for (int pass = 0; pass < 4; ++pass) {
        const int shift = 24 - 8 * pass;
        hist[t] = 0u;
        __syncthreads();

        #pragma unroll
        for (int j = 0; j < EPT; ++j) {
            bool part = (kbits == 0) || ((key[j] >> (32 - kbits)) == kprefix);
            if (part) atomicAdd(&hist[(key[j] >> shift) & 0xFFu], 1u);  // ds_add_u32
        }
        __syncthreads();

        // Wave 0 only: suffix-scan 256 bins (8 bins/lane) + pick digit. No barriers.
        if (t < 32) {
            uint4 h0 = *(const uint4*)&hist[t * 8];
            uint4 h1 = *(const uint4*)&hist[t * 8 + 4];
            uint32_t c[8] = {h0.x, h0.y, h0.z, h0.w, h1.x, h1.y, h1.z, h1.w};
            uint32_t s[8];
            s[7] = c[7];
            #pragma unroll
            for (int j = 6; j >= 0; --j) s[j] = c[j] + s[j + 1];
            uint32_t inc = s[0];                       // my 8-bin total
            #pragma unroll
            for (int off = 1; off < 32; off <<= 1) {   // wave suffix scan
                uint32_t o = __shfl_down(inc, off, 32);
                if (t + off < 32) inc += o;
            }
            uint32_t excl = inc - s[0];                // count of digits >= (t+1)*8
            #pragma unroll
            for (int j = 0; j < 8; ++j) {
                uint32_t ge  = s[j] + excl;                          // cnt_ge(d)
                uint32_t geN = (j < 7) ? (s[j + 1] + excl) : excl;   // cnt_ge(d+1)
                if (ge >= rem && geN < rem) {
                    s_digit = (uint32_t)(t * 8 + j);
                    s_rem   = rem - geN;
                }
            }
        }
        __syncthreads();
        kprefix = (kprefix << 8) | s_digit;
        rem     = s_rem;
        kbits  += 8;
    }

    const uint32_t T = kprefix;  // exact key of the k-th largest element
    // Keys > T: all kept. Keys == T: keep first 'rem' by ascending index (jax tie rule).

    // ---- Index-ordered rank among threshold-equal elements ----
    uint32_t ceq = 0;
    #pragma unroll
    for (int j = 0; j < EPT; ++j) ceq += (key[j] == T) ? 1u : 0u;
    hist[t] = ceq;
    __syncthreads();

    if (t < 32) {  // wave 0: exclusive prefix over 256 thread counts
        uint4 h0 = *(const uint4*)&hist[t * 8];
        uint4 h1 = *(const uint4*)&hist[t * 8 + 4];
        uint32_t c[8] = {h0.x, h0.y, h0.z, h0.w, h1.x, h1.y, h1.z, h1.w};
        uint32_t e[8];
        e[0] = 0;
        #pragma unroll
        for (int j = 1; j < 8; ++j) e[j] = e[j - 1] + c[j - 1];
        uint32_t tot = e[7] + c[7];
        uint32_t inc = tot;
        #pragma unroll
        for (int off = 1; off < 32; off <<= 1) {   // wave prefix scan
            uint32_t o = __shfl_up(inc, off, 32);
            if (t >= off) inc += o;
        }
        uint32_t exclw = inc - tot;                // equals in lanes < t
        uint4 p0 = make_uint4(exclw + e[0], exclw + e[1], exclw + e[2], exclw + e[3]);
        uint4 p1 = make_uint4(exclw + e[4], exclw + e[5], exclw + e[6], exclw + e[7]);
        *(uint4*)&pfx[t * 8]     = p0;
        *(uint4*)&pfx[t * 8 + 4] = p1;
    }
    __syncthreads();
    uint32_t eqrank = pfx[t];  // equal-to-threshold elements at lower indices

    // ---- Emit masked row from registers (2x global_store_b128) ----
    float res[EPT];
    #pragma unroll
    for (int j = 0; j < EPT; ++j) {
        float o = 0.0f;
        if (key[j] > T) {
            o = f[j];
        } else if (key[j] == T) {
            if (eqrank < rem) o = f[j];
            eqrank++;
        }
        res[j] = o;
    }
    float4* o4 = (float4*)(orow + base);
    o4[0] = make_float4(res[0], res[1], res[2], res[3]);
    o4[1] = make_float4(res[4], res[5], res[6], res[7]);
}

extern "C" void kernel_launch(void* const* d_in, const int* in_sizes, int n_in,
                              void* d_out, int out_size, void* d_ws, size_t ws_size,
                              hipStream_t stream) {
    const float* x  = (const float*)d_in[0];
    const int*   kp = (const int*)d_in[1];
    float*       o  = (float*)d_out;
    const int rows  = in_sizes[0] / DDIM;  // B*S = 32768
    topk_sparsify_kernel<<<rows, TPB, 0, stream>>>(x, kp, o);
}